// Decoder_14199161880853
// MI455X (gfx1250) — compile-verified
//
#include <hip/hip_runtime.h>
#include <hip/hip_bf16.h>

// ---------------------------------------------------------------------------
// Decoder step for MI455X (gfx1250, wave32).
// f32 path via V_WMMA_F32_16X16X4_F32 (16x16 C tile, K=4 per instruction).
// Adds: GLOBAL_LOAD_ASYNC_TO_LDS_B128 staging (ASYNCcnt) + global_prefetch_b8.
// ---------------------------------------------------------------------------

typedef __attribute__((ext_vector_type(2))) float v2f;
typedef __attribute__((ext_vector_type(8))) float v8f;
typedef __attribute__((ext_vector_type(4))) int   v4i;

#define Vv 32000
#define Ee 512
#define Hh 1024
#define Aa 512
#define Bb 64
#define Ss 128
#define H2 2048     // 2H
#define H4 4096     // 4H
#define H3 3072     // 3H
#define X0W 2560    // E + 2H

// d_out layout (floats)
#define OUT_LOGITS 0
#define OUT_HID    (Bb * Vv)                 // 2048000
#define OUT_CELL   (OUT_HID + 2 * Bb * Hh)   // 2179072
#define OUT_ALPHA  (OUT_CELL + 2 * Bb * Hh)  // 2310144

// workspace layout (floats)
#define WS_X0    0                           // (B, 2560)
#define WS_DEC   (WS_X0 + Bb * X0W)          // (B, 512)
#define WS_SCORE (WS_DEC + Bb * Aa)          // (S*B) score_t[s*B+b]
#define WS_GATES (WS_SCORE + Ss * Bb)        // (B, 4096)
#define WS_H0    (WS_GATES + Bb * H4)        // (B, 1024)
#define WS_Z     (WS_H0 + Bb * Hh)           // (B, 3072)

#if defined(__gfx1250__) && __has_builtin(__builtin_amdgcn_global_load_async_to_lds_b128)
#define USE_ASYNC_LDS 1
#else
#define USE_ASYNC_LDS 0
#endif

__device__ __forceinline__ v8f wmma_f32(v2f a, v2f b, v8f c) {
    // (neg_a, A, neg_b, B, c_mod, C, reuse_a, reuse_b)
    return __builtin_amdgcn_wmma_f32_16x16x4_f32(false, a, false, b, (short)0, c,
                                                 false, false);
}

__device__ __forceinline__ float sigf(float x) { return 1.0f / (1.0f + expf(-x)); }

// ------------------------- embedding gather --------------------------------
__global__ void embed_kernel(const int* __restrict__ token,
                             const float* __restrict__ emb, float* __restrict__ x0) {
    int b = blockIdx.x;
    int t = token[b];
    for (int e = threadIdx.x; e < Ee; e += blockDim.x)
        x0[b * X0W + e] = emb[t * Ee + e];
}

// --------------- generic WMMA GEMM: C = A1*W1^T (+ A2*W2^T) + bias ---------
// A row-major (M,K), W row-major (N,K), C row-major (M,N). One 16x16 tile/wave.
// K must be a multiple of 64 (true for 1024/2560/3072).
__global__ void gemm_wmma(const float* __restrict__ A1, const float* __restrict__ W1, int K1,
                          const float* __restrict__ A2, const float* __restrict__ W2, int K2,
                          const float* __restrict__ bias1, const float* __restrict__ bias2,
                          float* __restrict__ C, int M, int N) {
    int wavesPerBlock = blockDim.x >> 5;
    int tile = blockIdx.x * wavesPerBlock + (threadIdx.x >> 5);
    int ntiles = (M >> 4) * (N >> 4);
    if (tile >= ntiles) return;                 // wave-uniform
    int nt = N >> 4;
    int r0 = (tile / nt) << 4;
    int n0 = (tile % nt) << 4;

    int lane = threadIdx.x & 31;
    int half = lane >> 4;                       // 0: lanes 0-15, 1: lanes 16-31
    int mr   = lane & 15;                       // A row within tile
    int nn   = lane & 15;                       // C/B column within tile

    v8f acc = {0.f, 0.f, 0.f, 0.f, 0.f, 0.f, 0.f, 0.f};

    const float* a1row = A1 + (r0 + mr) * K1 + 2 * half;
    const float* w1row = W1 + (n0 + nn) * K1 + 2 * half;
    for (int k = 0; k < K1; k += 64) {
        __builtin_prefetch(w1row + k + 256, 0, 1);   // global_prefetch_b8, 1KB ahead
#pragma unroll
        for (int kk = 0; kk < 64; kk += 4) {
            v2f a = *(const v2f*)(a1row + k + kk);
            v2f b = *(const v2f*)(w1row + k + kk);
            acc = wmma_f32(a, b, acc);
        }
    }
    if (A2) {
        const float* a2row = A2 + (r0 + mr) * K2 + 2 * half;
        const float* w2row = W2 + (n0 + nn) * K2 + 2 * half;
        for (int k = 0; k < K2; k += 64) {
            __builtin_prefetch(w2row + k + 256, 0, 1);
#pragma unroll
            for (int kk = 0; kk < 64; kk += 4) {
                v2f a = *(const v2f*)(a2row + k + kk);
                v2f b = *(const v2f*)(w2row + k + kk);
                acc = wmma_f32(a, b, acc);
            }
        }
    }
    float bsum = 0.f;
    if (bias1) bsum += bias1[n0 + nn];
    if (bias2) bsum += bias2[n0 + nn];

    // C layout: VGPR i holds row M=i (lanes 0-15) / M=i+8 (lanes 16-31)
#pragma unroll
    for (int i = 0; i < 8; ++i)
        C[(r0 + i + 8 * half) * N + n0 + nn] = acc[i] + bsum;
}

// ---------------- attention energy + score (tanh, v-dot) -------------------
// Rows r' = s*B + b (memory order of enc_out). Each block: one 16-row panel,
// all 32 column tiles of A=512. enc panel staged through LDS in 512-col chunks
// via GLOBAL_LOAD_ASYNC_TO_LDS_B128 (ASYNCcnt) when available.
#define APAD 516
__global__ void energy_score_kernel(const float* __restrict__ enc,     // (S,B,2H)
                                    const float* __restrict__ Wenc,    // (A,2H)
                                    const float* __restrict__ decp,    // (B,A)
                                    const float* __restrict__ vw,      // (A)
                                    float* __restrict__ score_t) {     // (S*B)
    __shared__ __align__(16) float As[16 * APAD];   // 16 x 512 chunk, padded
    __shared__ float sacc[16];

    int r0 = blockIdx.x << 4;         // 16 | 64  =>  panel never crosses s
    int b0 = r0 & (Bb - 1);
    int tid = threadIdx.x;
    int wave = tid >> 5;
    int lane = tid & 31;
    int half = lane >> 4;
    int mr   = lane & 15;
    int nn   = lane & 15;

    if (tid < 16) sacc[tid] = 0.f;

    v8f acc[4];
#pragma unroll
    for (int t = 0; t < 4; ++t) acc[t] = (v8f){0.f,0.f,0.f,0.f,0.f,0.f,0.f,0.f};

    for (int ch = 0; ch < 4; ++ch) {
        __syncthreads();
#if USE_ASYNC_LDS
        // 16 x 512 floats = 2048 x 16B chunks; 8 per thread, coalesced.
#pragma unroll
        for (int it = 0; it < 8; ++it) {
            int j = tid + 256 * it;
            int m = j >> 7, c4 = j & 127;
            const float* g = enc + (r0 + m) * H2 + ch * 512 + (c4 << 2);
            float* l = &As[m * APAD + (c4 << 2)];
            __builtin_amdgcn_global_load_async_to_lds_b128((v4i*)g, (v4i*)l, 0, 0);
        }
        asm volatile("s_wait_asynccnt 0" ::: "memory");
#else
        for (int j = tid; j < 16 * 512; j += 256) {
            int m = j >> 9, c = j & 511;
            As[m * APAD + c] = enc[(r0 + m) * H2 + ch * 512 + c];
        }
#endif
        __syncthreads();
#pragma unroll
        for (int t = 0; t < 4; ++t) {
            int n0 = ((wave << 2) + t) << 4;
            const float* wrow = Wenc + (n0 + nn) * H2 + ch * 512 + 2 * half;
            const float* arow = &As[mr * APAD + 2 * half];
            for (int k = 0; k < 512; k += 64) {
                __builtin_prefetch(wrow + k + 256, 0, 1);
#pragma unroll
                for (int kk = 0; kk < 64; kk += 4) {
                    v2f a = *(const v2f*)(arow + k + kk);   // ds_load_b64
                    v2f b = *(const v2f*)(wrow + k + kk);
                    acc[t] = wmma_f32(a, b, acc[t]);
                }
            }
        }
    }
    __syncthreads();

    // epilogue: e = tanh(c + dec_proj[b][n]); score_m += e * v[n]
#pragma unroll
    for (int t = 0; t < 4; ++t) {
        int n0 = ((wave << 2) + t) << 4;
        int n = n0 + nn;
        float v = vw[n];
#pragma unroll
        for (int i = 0; i < 8; ++i) {
            int m = i + 8 * half;
            float e = tanhf(acc[t][i] + decp[(b0 + m) * Aa + n]);
            atomicAdd(&sacc[m], e * v);           // ds_add_f32
        }
    }
    __syncthreads();
    if (tid < 16) score_t[r0 + tid] = sacc[tid];
}

// ---------------- softmax over S, context = alpha . enc --------------------
__global__ void softmax_context_kernel(const float* __restrict__ score_t,  // (S*B)
                                       const float* __restrict__ enc,      // (S,B,2H)
                                       float* __restrict__ alpha_out,      // (B,S)
                                       float* __restrict__ x0,             // (B,2560)
                                       float* __restrict__ z) {            // (B,3072)
    __shared__ float red[Ss];
    __shared__ float alf[Ss];
    int b = blockIdx.x;
    int s = threadIdx.x;                          // blockDim.x == 128 == S

    float sc = score_t[s * Bb + b];
    red[s] = sc; __syncthreads();
    for (int off = 64; off > 0; off >>= 1) {
        if (s < off) red[s] = fmaxf(red[s], red[s + off]);
        __syncthreads();
    }
    float mx = red[0]; __syncthreads();
    float e = expf(sc - mx);
    red[s] = e; __syncthreads();
    for (int off = 64; off > 0; off >>= 1) {
        if (s < off) red[s] += red[s + off];
        __syncthreads();
    }
    float a = e / red[0];
    alf[s] = a;
    alpha_out[b * Ss + s] = a;
    __syncthreads();

    for (int d = s; d < H2; d += Ss) {
        float acc = 0.f;
#pragma unroll 4
        for (int k = 0; k < Ss; ++k)
            acc += alf[k] * enc[k * (Bb * H2) + b * H2 + d];
        x0[b * X0W + Ee + d] = acc;
        z[b * H3 + Hh + d]   = acc;
    }
}

// ---------------------------- LSTM pointwise -------------------------------
__global__ void lstm_pointwise(const float* __restrict__ gates,   // (B,4H)
                               const float* __restrict__ cprev,   // (B,H)
                               float* __restrict__ hA,            // (B,H)
                               float* __restrict__ hB, int ldb,   // optional copy
                               float* __restrict__ cOut) {        // (B,H)
    int idx = blockIdx.x * blockDim.x + threadIdx.x;
    if (idx >= Bb * Hh) return;
    int b = idx >> 10, n = idx & (Hh - 1);
    const float* g = gates + b * H4;
    float i = sigf(g[n]);
    float f = sigf(g[n + Hh]);
    float gg = tanhf(g[n + 2 * Hh]);
    float o = sigf(g[n + 3 * Hh]);
    float c = f * cprev[b * Hh + n] + i * gg;
    float h = o * tanhf(c);
    hA[b * Hh + n] = h;
    if (hB) hB[b * ldb + n] = h;
    cOut[b * Hh + n] = c;
}

// ---------------------------------------------------------------------------
extern "C" void kernel_launch(void* const* d_in, const int* in_sizes, int n_in,
                              void* d_out, int out_size, void* d_ws, size_t ws_size,
                              hipStream_t stream) {
    const int*   token  = (const int*)d_in[0];
    const float* hidden = (const float*)d_in[1];
    const float* cell   = (const float*)d_in[2];
    const float* enc    = (const float*)d_in[3];
    const float* emb    = (const float*)d_in[4];
    const float* W_enc  = (const float*)d_in[5];
    const float* W_dec  = (const float*)d_in[6];
    const float* v_w    = (const float*)d_in[7];
    const float* W_ih0  = (const float*)d_in[8];
    const float* W_hh0  = (const float*)d_in[9];
    const float* b_ih0  = (const float*)d_in[10];
    const float* b_hh0  = (const float*)d_in[11];
    const float* W_ih1  = (const float*)d_in[12];
    const float* W_hh1  = (const float*)d_in[13];
    const float* b_ih1  = (const float*)d_in[14];
    const float* b_hh1  = (const float*)d_in[15];
    const float* fc_W   = (const float*)d_in[16];
    const float* fc_b   = (const float*)d_in[17];

    float* out = (float*)d_out;
    float* ws  = (float*)d_ws;

    float* x0      = ws + WS_X0;
    float* dec_p   = ws + WS_DEC;
    float* score_t = ws + WS_SCORE;
    float* gates   = ws + WS_GATES;
    float* h0      = ws + WS_H0;
    float* z       = ws + WS_Z;

    float* o_logits = out + OUT_LOGITS;
    float* o_hid0   = out + OUT_HID;
    float* o_hid1   = out + OUT_HID + Bb * Hh;
    float* o_cel0   = out + OUT_CELL;
    float* o_cel1   = out + OUT_CELL + Bb * Hh;
    float* o_alpha  = out + OUT_ALPHA;

    const float* top  = hidden + Bb * Hh;   // hidden[-1] == layer 1
    const float* hid0 = hidden;
    const float* hid1 = hidden + Bb * Hh;
    const float* cel0 = cell;
    const float* cel1 = cell + Bb * Hh;

    // 1) embedding -> x0[:, :E]
    embed_kernel<<<Bb, 128, 0, stream>>>(token, emb, x0);

    // 2) dec_proj = top @ W_dec^T   (64x512, K=1024): 128 tiles / 4 waves
    gemm_wmma<<<32, 128, 0, stream>>>(top, W_dec, Hh, nullptr, nullptr, 0,
                                      nullptr, nullptr, dec_p, Bb, Aa);

    // 3) energy + score (8192x512, K=2048), fused tanh + v-dot
    energy_score_kernel<<<(Ss * Bb) / 16, 256, 0, stream>>>(enc, W_enc, dec_p, v_w,
                                                            score_t);

    // 4) softmax over S + context -> alpha(d_out), x0[:,E:], z[:,H:]
    softmax_context_kernel<<<Bb, Ss, 0, stream>>>(score_t, enc, o_alpha, x0, z);

    // 5) gates0 = x0 @ W_ih0^T + hid0 @ W_hh0^T + b   (64x4096): 1024 tiles
    gemm_wmma<<<256, 128, 0, stream>>>(x0, W_ih0, X0W, hid0, W_hh0, Hh,
                                       b_ih0, b_hh0, gates, Bb, H4);

    // 6) LSTM cell 0 -> h0 (ws + d_out), c0 (d_out)
    lstm_pointwise<<<(Bb * Hh) / 256, 256, 0, stream>>>(gates, cel0, o_hid0,
                                                        h0, Hh, o_cel0);

    // 7) gates1 = h0 @ W_ih1^T + hid1 @ W_hh1^T + b   (64x4096)
    gemm_wmma<<<256, 128, 0, stream>>>(h0, W_ih1, Hh, hid1, W_hh1, Hh,
                                       b_ih1, b_hh1, gates, Bb, H4);

    // 8) LSTM cell 1 -> h1 into d_out and z[:, :H], c1 -> d_out
    lstm_pointwise<<<(Bb * Hh) / 256, 256, 0, stream>>>(gates, cel1, o_hid1,
                                                        z, H3, o_cel1);

    // 9) logits = z @ fc_W^T + fc_b   (64x32000, K=3072): 8000 tiles
    gemm_wmma<<<2000, 128, 0, stream>>>(z, fc_W, H3, nullptr, nullptr, 0,
                                        fc_b, nullptr, o_logits, Bb, Vv);
}